// FeedForwardQuantum_65481071396775
// MI455X (gfx1250) — compile-verified
//
#include <hip/hip_runtime.h>
#include <math.h>

#define NQ     8
#define EMBED  1024
#define WAVES  2                 // waves per block
#define NTHR   (WAVES * 32)

typedef float v2f __attribute__((ext_vector_type(2)));
typedef float v4f __attribute__((ext_vector_type(4)));
typedef float v8f __attribute__((ext_vector_type(8)));
typedef int   v4i __attribute__((ext_vector_type(4)));

#if defined(__gfx1250__) && __has_builtin(__builtin_amdgcn_global_load_async_to_lds_b128)
#define HAVE_ASYNC_LDS 1
#else
#define HAVE_ASYNC_LDS 0
#endif

#if HAVE_ASYNC_LDS
typedef __attribute__((address_space(1))) v4i* gv4i_p;   // global int4*
typedef __attribute__((address_space(3))) v4i* lv4i_p;   // LDS int4*
#endif

__device__ __forceinline__ float wave_sum32(float v) {
#pragma unroll
  for (int off = 16; off > 0; off >>= 1)
    v += __shfl_xor(v, off, 32);
  return v;
}

// Stage NV4 float4s from global into LDS; async data-mover path when available.
template <int NV4>
__device__ __forceinline__ void stage_lds(float* dst, const float* __restrict__ src, int tid) {
#if HAVE_ASYNC_LDS
#pragma unroll 1
  for (int i = tid; i < NV4; i += NTHR)
    __builtin_amdgcn_global_load_async_to_lds_b128(
        (gv4i_p)(src + 4 * i), (lv4i_p)(dst + 4 * i), 0, 0);
#else
#pragma unroll 1
  for (int i = tid; i < NV4; i += NTHR)
    ((v4f*)dst)[i] = ((const v4f*)src)[i];
#endif
}

__device__ __forceinline__ void wait_stage() {
#if HAVE_ASYNC_LDS
#if __has_builtin(__builtin_amdgcn_s_wait_asynccnt)
  __builtin_amdgcn_s_wait_asynccnt(0);
#else
  asm volatile("s_wait_asynccnt 0" ::: "memory");
#endif
#endif
}

// Fused: q_in = x@W1[0:8].T + b1[0:8]  ->  8-qubit sim  ->  out = q_out@W2.T + b2
__global__ __launch_bounds__(NTHR) void ffq_fused(
    const float* __restrict__ x,  const float* __restrict__ W1,
    const float* __restrict__ b1, const float* __restrict__ qp,
    const float* __restrict__ W2, const float* __restrict__ b2,
    float* __restrict__ out, int tokens)
{
  __shared__ __align__(16) float lds_mat[NQ * EMBED];  // W1[0:8][:], later W2 (1024x8)
  __shared__ __align__(16) float lds_b2[EMBED];
  __shared__ float lds_gate[NQ][8];                    // Rot gates g00..g11 (re,im)
  __shared__ float lds_qout[WAVES][16][NQ];            // per-wave 16x8 q_out

  const int tid  = threadIdx.x;
  const int lane = tid & 31;
  const int wave = tid >> 5;
  const int strip = blockIdx.x * WAVES + wave;         // 16-row strip id
  const int row0  = strip * 16;

  // ---- stage W1 rows 0..7 (32KB, contiguous) into LDS (async path) ----
  stage_lds<NQ * EMBED / 4>(lds_mat, W1, tid);

  // ---- shared Rot gates from qparams ----
  if (tid < NQ) {
    float phi = qp[tid*3+0], th = qp[tid*3+1], om = qp[tid*3+2];
    float ct = cosf(0.5f*th), st = sinf(0.5f*th);
    float s  = 0.5f*(phi+om), d = 0.5f*(phi-om);
    float epr = cosf(s), epi = -sinf(s);                // exp(-i(phi+om)/2)
    float emr = cosf(d), emi =  sinf(d);                // exp(+i(phi-om)/2)
    lds_gate[tid][0] =  epr*ct; lds_gate[tid][1] =  epi*ct;   // g00 = ep*ct
    lds_gate[tid][2] = -emr*st; lds_gate[tid][3] = -emi*st;   // g01 = -em*st
    lds_gate[tid][4] =  emr*st; lds_gate[tid][5] = -emi*st;   // g10 = conj(em)*st
    lds_gate[tid][6] =  epr*ct; lds_gate[tid][7] = -epi*ct;   // g11 = conj(ep)*ct
  }
  wait_stage();
  __syncthreads();

  float b1v[NQ];
#pragma unroll
  for (int q = 0; q < NQ; ++q) b1v[q] = b1[q];

  // ================= per-row: skinny GEMM1 + quantum sim =================
  for (int r = 0; r < 16; ++r) {
    const int row = row0 + r;
    const v4f* xrow4 = (const v4f*)(x + (size_t)row * EMBED);
    const v4f* w14   = (const v4f*)lds_mat;

    // q_in[q] = sum_k x[row][k] * W1[q][k]  (b128 streaming x reads, LDS W1)
    float acc[NQ];
#pragma unroll
    for (int q = 0; q < NQ; ++q) acc[q] = 0.f;
#pragma unroll 2
    for (int kc = 0; kc < EMBED / 128; ++kc) {
      v4f xv = __builtin_nontemporal_load(&xrow4[kc * 32 + lane]);
#pragma unroll
      for (int q = 0; q < NQ; ++q) {
        v4f wv = w14[q * 256 + kc * 32 + lane];
        acc[q] = fmaf(xv.x, wv.x, acc[q]);
        acc[q] = fmaf(xv.y, wv.y, acc[q]);
        acc[q] = fmaf(xv.z, wv.z, acc[q]);
        acc[q] = fmaf(xv.w, wv.w, acc[q]);
      }
    }
    float qv[NQ];
#pragma unroll
    for (int q = 0; q < NQ; ++q) qv[q] = wave_sum32(acc[q]) + b1v[q];

    // ---- state: 256 amps, idx = j*32 + lane; wire w acts on bit (7-w) ----
    float sr[8], si[8];
#pragma unroll
    for (int j = 0; j < 8; ++j) { sr[j] = 0.f; si[j] = 0.f; }
    if (lane == 0) sr[0] = 1.f;

    // RX(q_in[w]) layer: gate [[c, -i t],[-i t, c]], t = sin(half)
#pragma unroll
    for (int w = 0; w < 8; ++w) {
      const int b = 7 - w;
      float c = cosf(0.5f * qv[w]);
      float t = sinf(0.5f * qv[w]);
      if (b >= 5) {
        const int jb = 1 << (b - 5);
#pragma unroll
        for (int j = 0; j < 8; ++j) {
          if ((j & jb) == 0) {
            const int j2 = j | jb;
            float ar = sr[j], ai = si[j], br = sr[j2], bi = si[j2];
            sr[j]  = c*ar + t*bi;  si[j]  = c*ai - t*br;
            sr[j2] = c*br + t*ai;  si[j2] = c*bi - t*ar;
          }
        }
      } else {
        const int m = 1 << b;
#pragma unroll
        for (int j = 0; j < 8; ++j) {
          float pr = __shfl_xor(sr[j], m, 32);
          float pi = __shfl_xor(si[j], m, 32);
          float nr = c*sr[j] + t*pi;
          float ni = c*si[j] - t*pr;
          sr[j] = nr; si[j] = ni;
        }
      }
    }

    // Rot(phi,theta,omega) layer: general complex 2x2 from LDS
#pragma unroll
    for (int w = 0; w < 8; ++w) {
      const int b = 7 - w;
      float g00r = lds_gate[w][0], g00i = lds_gate[w][1];
      float g01r = lds_gate[w][2], g01i = lds_gate[w][3];
      float g10r = lds_gate[w][4], g10i = lds_gate[w][5];
      float g11r = lds_gate[w][6], g11i = lds_gate[w][7];
      if (b >= 5) {
        const int jb = 1 << (b - 5);
#pragma unroll
        for (int j = 0; j < 8; ++j) {
          if ((j & jb) == 0) {
            const int j2 = j | jb;
            float ar = sr[j], ai = si[j], br = sr[j2], bi = si[j2];
            sr[j]  = g00r*ar - g00i*ai + g01r*br - g01i*bi;
            si[j]  = g00r*ai + g00i*ar + g01r*bi + g01i*br;
            sr[j2] = g10r*ar - g10i*ai + g11r*br - g11i*bi;
            si[j2] = g10r*ai + g10i*ar + g11r*bi + g11i*br;
          }
        }
      } else {
        const int m = 1 << b;
        const bool hi = (lane & m) != 0;
        float car = hi ? g11r : g00r, cai = hi ? g11i : g00i;  // * own amp
        float cbr = hi ? g10r : g01r, cbi = hi ? g10i : g01i;  // * partner
#pragma unroll
        for (int j = 0; j < 8; ++j) {
          float pr = __shfl_xor(sr[j], m, 32);
          float pi = __shfl_xor(si[j], m, 32);
          float ar = sr[j], ai = si[j];
          sr[j] = car*ar - cai*ai + cbr*pr - cbi*pi;
          si[j] = car*ai + cai*ar + cbr*pi + cbi*pr;
        }
      }
    }

    // CNOT ring is a basis permutation: fold into readout signs (no data motion)
    float z[NQ];
#pragma unroll
    for (int i = 0; i < NQ; ++i) z[i] = 0.f;
#pragma unroll
    for (int j = 0; j < 8; ++j) {
      float p = sr[j]*sr[j] + si[j]*si[j];
      int idx = (j << 5) | lane;
#pragma unroll
      for (int w = 0; w < 7; ++w)                 // CNOT(w, w+1)
        if ((idx >> (7 - w)) & 1) idx ^= 1 << (6 - w);
      if (idx & 1) idx ^= 128;                    // CNOT(7, 0)
#pragma unroll
      for (int i = 0; i < NQ; ++i)
        z[i] += ((idx >> (7 - i)) & 1) ? -p : p;
    }
#pragma unroll
    for (int i = 0; i < NQ; ++i) {
      float zi = wave_sum32(z[i]);
      if (lane == i) lds_qout[wave][r][i] = zi;
    }
  }

  // ---- swap LDS matrix buffer: W1 -> W2 (1024x8, 32KB) + b2 ----
  __syncthreads();
  stage_lds<NQ * EMBED / 4>(lds_mat, W2, tid);
  stage_lds<EMBED / 4>(lds_b2, b2, tid);
  wait_stage();
  __syncthreads();

  // ================= GEMM2: out[16x1024] = q_out[16x8] @ W2.T + b2 =================
  const int mrow  = lane & 15;              // A: M index / B,C: N index
  const int khalf = (lane < 16) ? 0 : 2;    // f32 16x16x4 operand K split

#if __has_builtin(__builtin_amdgcn_wmma_f32_16x16x4_f32)
  v2f a_lo, a_hi;
  a_lo.x = lds_qout[wave][mrow][khalf + 0];
  a_lo.y = lds_qout[wave][mrow][khalf + 1];
  a_hi.x = lds_qout[wave][mrow][4 + khalf + 0];
  a_hi.y = lds_qout[wave][mrow][4 + khalf + 1];

  for (int nt = 0; nt < EMBED / 16; ++nt) {
    const int col = nt * 16 + mrow;
    v2f b_lo, b_hi;
    b_lo.x = lds_mat[col * 8 + khalf + 0];
    b_lo.y = lds_mat[col * 8 + khalf + 1];
    b_hi.x = lds_mat[col * 8 + 4 + khalf + 0];
    b_hi.y = lds_mat[col * 8 + 4 + khalf + 1];

    v8f c = {};
    c = __builtin_amdgcn_wmma_f32_16x16x4_f32(false, a_lo, false, b_lo,
                                              (short)0, c, false, false);
    c = __builtin_amdgcn_wmma_f32_16x16x4_f32(false, a_hi, false, b_hi,
                                              (short)0, c, false, false);
    float badd = lds_b2[col];
#pragma unroll
    for (int v = 0; v < 8; ++v) {
      const int row = row0 + v + ((lane >> 4) << 3);   // C layout: M = v + 8*(lane/16)
      __builtin_nontemporal_store(c[v] + badd, &out[(size_t)row * EMBED + col]);
    }
  }
#else
  // VALU fallback (should not be taken on gfx1250)
  for (int nt = 0; nt < EMBED / 16; ++nt) {
    const int col = nt * 16 + mrow;
    float badd = lds_b2[col];
#pragma unroll
    for (int v = 0; v < 8; ++v) {
      const int M = v + ((lane >> 4) << 3);
      float s = 0.f;
#pragma unroll
      for (int k = 0; k < NQ; ++k)
        s = fmaf(lds_qout[wave][M][k], lds_mat[col * 8 + k], s);
      __builtin_nontemporal_store(s + badd, &out[(size_t)(row0 + M) * EMBED + col]);
    }
  }
#endif
  (void)tokens;
}

extern "C" void kernel_launch(void* const* d_in, const int* in_sizes, int n_in,
                              void* d_out, int out_size, void* d_ws, size_t ws_size,
                              hipStream_t stream) {
  (void)n_in; (void)out_size; (void)d_ws; (void)ws_size;
  const float* x  = (const float*)d_in[0];
  const float* W1 = (const float*)d_in[1];
  const float* b1 = (const float*)d_in[2];
  const float* qp = (const float*)d_in[3];
  const float* W2 = (const float*)d_in[4];
  const float* b2 = (const float*)d_in[5];
  float* out = (float*)d_out;

  const int tokens = in_sizes[0] / EMBED;        // 8192
  const int blocks = tokens / (16 * WAVES);      // 16 rows/wave -> 256 blocks
  ffq_fused<<<blocks, NTHR, 0, stream>>>(x, W1, b1, qp, W2, b2, out, tokens);
}